// S2VTransformer_34110630265654
// MI455X (gfx1250) — compile-verified
//
#include <hip/hip_runtime.h>

typedef __attribute__((ext_vector_type(2))) float v2f;
typedef __attribute__((ext_vector_type(8))) float v8f;

#define H_   128
#define W_   128
#define S_   72
#define SPS_ 24      // slices per stack
#define ZS_  120     // Z_SCALE (output depth)
#define B_   4

constexpr float ST_RATIO_ = 5.0f;
constexpr int GROUPS = S_ / 4;                     // 18 slice-groups of 4
constexpr int TILES_PER_WAVE = 16;                 // 16 pixels per tile
constexpr int CHUNKS = (H_ * W_ / 16) / TILES_PER_WAVE; // 64 chunks per (b,g)
constexpr size_t ND = (size_t)B_ * H_ * W_ * ZS_;  // 7,864,320 per output tensor

// ---------------------------------------------------------------- zero d_out
__global__ __launch_bounds__(256) void s2v_zero(float4* __restrict__ out, int n4) {
    int idx = blockIdx.x * blockDim.x + threadIdx.x;
    if (idx < n4) out[idx] = make_float4(0.f, 0.f, 0.f, 0.f);
}

// ------------------------------------------------- WMMA transform + scatter
// One wave owns 4 consecutive slices (a "group") and a run of pixel tiles.
// A-matrix (16x4 f32): rows 0-2 = slice q0 (x,y,z), rows 3-5 = q1,
//                      rows 8-10 = q2, rows 11-13 = q3, rest zero.
// Row layout (folds orientation + through-plane t + z-pad into 'const'):
//   coord = coef_i * i + coef_j * j + const * 1 + 0
// B-matrix (4x16 f32): column n = (i_n, j_n, 1, 0) for 16 pixels.
// C layout puts slice q0/q1 in lanes 0-15 (c[0..2], c[3..5]) and q2/q3 in
// lanes 16-31 (c[0..2], c[3..5]) -> each lane scatters 2 (slice,pixel) pairs.
__global__ __launch_bounds__(256) void s2v_scatter(
        const float* __restrict__ vol,   // (B,H,W,S,1)
        const float* __restrict__ trf,   // (B,S,3,4)
        float* __restrict__ Nout,        // (B,H,W,ZS,1)
        float* __restrict__ Dout) {      // (B,H,W,ZS,1)
    const int lane   = threadIdx.x & 31;
    const int waveId = blockIdx.x * (blockDim.x >> 5) + (threadIdx.x >> 5);
    const int chunk  = waveId % CHUNKS;
    const int g      = (waveId / CHUNKS) % GROUPS;
    const int b      = waveId / (CHUNKS * GROUPS);

    // ---- build the wave-invariant A fragment ----
    const int r = lane & 15;               // matrix row this lane contributes to
    int q = -1, axis = 0;
    if (r < 6)                { q = r / 3;           axis = r % 3; }
    else if (r >= 8 && r < 14){ q = 2 + (r - 8) / 3; axis = (r - 8) % 3; }

    v2f a_frag = {0.f, 0.f};
    if (q >= 0) {
        const int   s = g * 4 + q;
        const int   o = s / SPS_;                              // 0=axi,1=cor,2=sag
        const float t = (float)(s % SPS_) * ST_RATIO_;
        const float* Ar = trf + (((size_t)b * S_ + s) * 3 + axis) * 4;
        const float A0 = Ar[0] + (axis == 0 ? 1.f : 0.f);      // A = trf + eye
        const float A1 = Ar[1] + (axis == 1 ? 1.f : 0.f);
        const float A2 = Ar[2] + (axis == 2 ? 1.f : 0.f);
        const float A3 = Ar[3];
        float ci, cj, cc;
        if (o == 0)      { ci = A0; cj = A1; cc = A2 * t + A3; }  // base=(i,j,t,1)
        else if (o == 1) { ci = A0; cj = A2; cc = A1 * t + A3; }  // base=(i,t,j,1)
        else             { ci = A1; cj = A2; cc = A0 * t + A3; }  // base=(t,i,j,1)
        if (axis == 2) cc += 1.0f;                                // z pad offset
        if (lane < 16) { a_frag[0] = ci; a_frag[1] = cj; }        // K=0,1
        else           { a_frag[0] = cc; a_frag[1] = 0.f; }       // K=2,3
    }

    const int sBase = g * 4 + ((lane >> 4) << 1);  // this lane's 2 slices

    for (int tt = 0; tt < TILES_PER_WAVE; ++tt) {
        const int p = (chunk * TILES_PER_WAVE + tt) * 16 + (lane & 15);
        const int i = p >> 7;          // p / W_
        const int j = p & (W_ - 1);    // p % W_

        v2f b_frag;
        if (lane < 16) { b_frag[0] = (float)i; b_frag[1] = (float)j; } // K=0,1
        else           { b_frag[0] = 1.f;      b_frag[1] = 0.f;      } // K=2,3

        v8f c_frag = {0.f, 0.f, 0.f, 0.f, 0.f, 0.f, 0.f, 0.f};
        c_frag = __builtin_amdgcn_wmma_f32_16x16x4_f32(
            /*neg_a=*/false, a_frag, /*neg_b=*/false, b_frag,
            /*c_mod=*/(short)0, c_frag, /*reuse_a=*/false, /*reuse_b=*/false);

#pragma unroll
        for (int h = 0; h < 2; ++h) {
            const int   s = sBase + h;
            const float x = c_frag[3 * h + 0];
            const float y = c_frag[3 * h + 1];
            const float z = c_frag[3 * h + 2];          // already has +1 pad
            const float v = vol[(((size_t)b * H_ + i) * W_ + j) * S_ + s];

            const float x0f = floorf(x), y0f = floorf(y), z0f = floorf(z);
            const float wx = x - x0f, wy = y - y0f, wz = z - z0f;
            const float wx1 = 1.f - wx, wy1 = 1.f - wy, wz1 = 1.f - wz;
            const int x0 = (int)x0f, y0 = (int)y0f, z0 = (int)z0f;

            // per-axis validity, hoisted out of the 8-way unroll
            const bool vx0 = (unsigned)x0       < (unsigned)H_;  // xi = x0
            const bool vx1 = (unsigned)(x0 + 1) < (unsigned)H_;  // xi = x0+1
            const bool vy0 = (unsigned)y0       < (unsigned)W_;
            const bool vy1 = (unsigned)(y0 + 1) < (unsigned)W_;
            const bool vz0 = (unsigned)(z0 - 1) < (unsigned)ZS_; // zi=z0   in [1,120]
            const bool vz1 = (unsigned)z0       < (unsigned)ZS_; // zi=z0+1 in [1,120]

            // xy weight products (shared between the two z planes)
            const float w00 = wx1 * wy1, w10 = wx * wy1;
            const float w01 = wx1 * wy,  w11 = wx * wy;

            // single 64-bit base; neighbor offsets are compile-time constants
            // that fold into GLOBAL_ATOMIC_ADD_F32's 24-bit IOFFSET.
            const long long base =
                (((long long)b * H_ + x0) * W_ + y0) * ZS_ + (z0 - 1);
            float* __restrict__ nb = Nout + base;
            float* __restrict__ db = Dout + base;

#pragma unroll
            for (int d8 = 0; d8 < 8; ++d8) {
                const int dx = d8 & 1, dy = (d8 >> 1) & 1, dz = (d8 >> 2) & 1;
                const bool ok = (dx ? vx1 : vx0) & (dy ? vy1 : vy0)
                              & (dz ? vz1 : vz0);
                if (ok) {
                    const float wxy = dx ? (dy ? w11 : w10) : (dy ? w01 : w00);
                    const float w   = wxy * (dz ? wz : wz1);
                    const int off   = dx * (W_ * ZS_) + dy * ZS_ + dz; // const
                    atomicAdd(nb + off, w * v);   // global_atomic_add_f32
                    atomicAdd(db + off, w);
                }
            }
        }
    }
}

// ------------------------------------------------------- D = (D<=0) ? 1 : D
__global__ __launch_bounds__(256) void s2v_fixd(float* __restrict__ D, size_t n) {
    size_t idx = (size_t)blockIdx.x * blockDim.x + threadIdx.x;
    if (idx < n) {
        float v = D[idx];
        D[idx] = (v <= 0.f) ? 1.f : v;
    }
}

extern "C" void kernel_launch(void* const* d_in, const int* in_sizes, int n_in,
                              void* d_out, int out_size, void* d_ws, size_t ws_size,
                              hipStream_t stream) {
    const float* vol = (const float*)d_in[0];   // (4,128,128,72,1) f32
    const float* trf = (const float*)d_in[1];   // (4,72,3,4) f32
    float* out  = (float*)d_out;                // [N | D], each ND floats
    float* Nout = out;
    float* Dout = out + ND;

    // 1) zero both output tensors (harness poisons d_out; must re-zero per call)
    const int n4 = (int)((2 * ND) / 4);
    s2v_zero<<<(n4 + 255) / 256, 256, 0, stream>>>((float4*)out, n4);

    // 2) WMMA transform + trilinear scatter
    const int totalWaves = B_ * GROUPS * CHUNKS;      // 4608
    const int blocks     = totalWaves / 8;            // 8 waves / 256-thread block
    s2v_scatter<<<blocks, 256, 0, stream>>>(vol, trf, Nout, Dout);

    // 3) denominator clamp
    s2v_fixd<<<(int)((ND + 255) / 256), 256, 0, stream>>>(Dout, ND);
}